// MultiscaleLLN_56367150793312
// MI455X (gfx1250) — compile-verified
//
#include <hip/hip_runtime.h>

// Multiscale local-luminance normalization, fused single-pass per 16x16 tile.
// Separable 13-tap Gaussian expressed as banded matmuls on V_WMMA_F32_16X16X4_F32.
// Gaussian band + prefix-sum tables are lane-resident, read via ds_bpermute.
// Memory-bound: ~267 MB HBM traffic -> ~11.5 us roofline on MI455X.

typedef float v2f __attribute__((ext_vector_type(2)));
typedef float v4f __attribute__((ext_vector_type(4)));
typedef float v8f __attribute__((ext_vector_type(8)));

#define WAVES_PER_BLOCK 8
#define LTILE   28                  // 16 + 2*6 halo
#define LSTRIDE 28                  // L tile row stride (floats)
#define HSTRIDE 28                  // transposed H buffer: [16 cols][28 rows]
#define LDS_PER_WAVE (LTILE * LSTRIDE + 16 * HSTRIDE)   // 784 + 448 = 1232 floats

#define N_PIX   87040
#define EPS     1e-3f

// Unscaled Gaussian taps exp(-(i-6)^2/18); the 1/(2*pi*sigma^2) factor cancels
// exactly in lum/dens, so it is omitted everywhere.
constexpr float GT[7] = {   // taps by distance-from-center 6..0
    0.13533528f, 0.24935221f, 0.41111229f, 0.60653066f,
    0.80073740f, 0.94595947f, 1.00000000f
};
// Prefix sums of the smallest taps: P[k] = sum of k outermost taps (one side).
constexpr float GP[7] = {
    0.0f, 0.13533528f, 0.38468750f, 0.79579979f,
    1.40233040f, 2.20306778f, 3.14902735f
};
#define GFULL 7.29805470f           // sum of all 13 taps

// table(d): lane-resident table broadcast; hardware wraps index mod 32.
__device__ __forceinline__ float lane_tab(float tab, int d) {
    return __int_as_float(
        __builtin_amdgcn_ds_bpermute((int)(((unsigned)d) << 2),
                                     __float_as_int(tab)));
}

__global__ __launch_bounds__(256)
void MultiscaleLLN_56367150793312_kernel(const float* __restrict__ x,
                                         float* __restrict__ out) {
    __shared__ float lds[WAVES_PER_BLOCK * LDS_PER_WAVE];

    const int tid  = threadIdx.x;
    const int wave = tid >> 5;
    const int lane = tid & 31;
    float* Lb = lds + wave * LDS_PER_WAVE;   // 28x28 zero-padded luminance tile
    float* Hb = Lb + LTILE * LSTRIDE;        // 16x28 transposed horizontal result

    // ---- map wave -> (level, batch, tile): all shifts, all scalar --------
    // per-level tiles (x128 batch): 32768, 8192, 2048, 512 (cum 32768/40960/43008/43520)
    int t = blockIdx.x * WAVES_PER_BLOCK + wave;
    int lvl, r;
    if      (t < 32768) { lvl = 0; r = t; }
    else if (t < 40960) { lvl = 1; r = t - 32768; }
    else if (t < 43008) { lvl = 2; r = t - 40960; }
    else                { lvl = 3; r = t - 43008; }
    const int W  = 256 >> lvl;               // H == W at every level
    const int ts = 4 - lvl;                  // log2(tiles per dimension)
    const int b  = r >> (2 * ts);
    const int q  = r & ((1 << (2 * ts)) - 1);
    const int ty = q >> ts;
    const int tx = q & ((1 << ts) - 1);
    const int off = (lvl == 0) ? 0 : (lvl == 1) ? 65536 : (lvl == 2) ? 81920 : 86016;
    const int y0 = ty << 4, x0 = tx << 4;
    const int base = (b * N_PIX + off) * 3;

    const int half = lane >> 4;     // 16-lane half
    const int m    = lane & 15;     // row (A) / column (B,D) index
    const int kb   = half * 2;      // K base for this half (f32 A/B layouts)

    // ---- lane-resident tables: taps (lanes 0..12), prefix sums (0..6) ----
    const int e = (lane > 6) ? (lane - 6) : (6 - lane);   // lanes 13..31 -> e>6 -> 0
    float gl = 0.0f;
    gl = (e == 6) ? GT[0] : gl;
    gl = (e == 5) ? GT[1] : gl;
    gl = (e == 4) ? GT[2] : gl;
    gl = (e == 3) ? GT[3] : gl;
    gl = (e == 2) ? GT[4] : gl;
    gl = (e == 1) ? GT[5] : gl;
    gl = (e == 0) ? GT[6] : gl;
    float pl = 0.0f;                 // prefix table, indexed 0..6 only
    pl = (lane == 1) ? GP[1] : pl;
    pl = (lane == 2) ? GP[2] : pl;
    pl = (lane == 3) ? GP[3] : pl;
    pl = (lane == 4) ? GP[4] : pl;
    pl = (lane == 5) ? GP[5] : pl;
    pl = (lane == 6) ? GP[6] : pl;

    // Same band values serve as horizontal-B and vertical-A fragments:
    // gband[j] = { g[4j+kb-m], g[4j+kb+1-m] }  (zero outside [0,12]; every
    // out-of-band index wraps onto a zero-holding lane 13..31)
    v2f gband[7];
#pragma unroll
    for (int j = 0; j < 7; ++j) {
        const int d0 = 4 * j + kb - m;
        gband[j].x = lane_tab(gl, d0);
        gband[j].y = lane_tab(gl, d0 + 1);
    }

    // ---- load zero-padded luminance tile into LDS -----------------------
    for (int idx = lane; idx < LTILE * LTILE; idx += 32) {
        const int ly = idx / LTILE;
        const int lx = idx - ly * LTILE;
        const int gy = y0 - 6 + ly;
        const int gx = x0 - 6 + lx;
        float v = 0.0f;
        if ((unsigned)gy < (unsigned)W && (unsigned)gx < (unsigned)W) {
            const float* px = x + base + (gy * W + gx) * 3;
            v = 0.2989f * px[0] + 0.587f * px[1] + 0.114f * px[2];
        }
        Lb[ly * LSTRIDE + lx] = v;
    }
    __syncthreads();

    // ---- horizontal conv: H(28x16) = L(28x28) x Gband(28x16) ------------
    // two overlapping 16-row chunks (padded-tile rows 0..15 and 12..27)
#pragma unroll
    for (int c = 0; c < 2; ++c) {
        const int rowbase = c * 12;
        const float* lrow = Lb + (rowbase + m) * LSTRIDE + kb;
        v2f af[7];                            // A[m][4j+kb..+1], contiguous b64
#pragma unroll
        for (int j = 0; j < 7; ++j)
            af[j] = *(const v2f*)(lrow + 4 * j);
        v8f acc = {};
#pragma unroll
        for (int j = 0; j < 7; ++j)
            acc = __builtin_amdgcn_wmma_f32_16x16x4_f32(
                false, af[j], false, gband[j], (short)0, acc, false, false);
        // D VGPR i -> chunk rows half*8+i..+7: contiguous in transposed Hb
        float* hdst = Hb + m * HSTRIDE + rowbase + half * 8;   // 16B aligned
        *(v4f*)(hdst)     = (v4f){acc[0], acc[1], acc[2], acc[3]};
        *(v4f*)(hdst + 4) = (v4f){acc[4], acc[5], acc[6], acc[7]};
    }
    __syncthreads();

    // ---- vertical conv: lum(16x16) = Gband(16x28) x H(28x16) ------------
    const float* hrow = Hb + m * HSTRIDE + kb;
    v2f hf[7];                                // B[4j+kb..+1][m], contiguous b64
#pragma unroll
    for (int j = 0; j < 7; ++j)
        hf[j] = *(const v2f*)(hrow + 4 * j);
    v8f lum = {};
#pragma unroll
    for (int j = 0; j < 7; ++j)
        lum = __builtin_amdgcn_wmma_f32_16x16x4_f32(
            false, gband[j], false, hf[j], (short)0, lum, false, false);

    // ---- closed-form density via prefix sums, then normalize ------------
    // S(t) = GFULL - P[max(6-t,0)] - P[max(t-W+7,0)]  (tap symmetry)
    const float Sw = 0.2989f + 0.587f + 0.114f;
    const int gx  = x0 + m;
    const int c1x = (6 - gx > 0) ? (6 - gx) : 0;
    const int c2x = (gx - W + 7 > 0) ? (gx - W + 7) : 0;
    const float Sx = GFULL - lane_tab(pl, c1x) - lane_tab(pl, c2x);
    const float SwSx  = Sw * Sx;
    const float eSwSx = EPS * SwSx;

    const int gy0 = y0 + half * 8;            // D VGPR i -> rows i / i+8
    int p = base + (gy0 * W + gx) * 3;
    const int pstep = 3 * W;
#pragma unroll
    for (int i = 0; i < 8; ++i) {
        const int gy  = gy0 + i;
        const int c1y = (6 - gy > 0) ? (6 - gy) : 0;
        const int c2y = (gy - W + 7 > 0) ? (gy - W + 7) : 0;
        const float Sy = GFULL - lane_tab(pl, c1y) - lane_tab(pl, c2y);
        // x/(lum/dens+eps) == x*dens/(lum + eps*dens): one division per row
        const float dens = SwSx * Sy;
        const float inv  = dens / (lum[i] + eSwSx * Sy);
        out[p]     = x[p]     * inv;          // re-read x: L2-resident
        out[p + 1] = x[p + 1] * inv;
        out[p + 2] = x[p + 2] * inv;
        p += pstep;
    }
}

extern "C" void kernel_launch(void* const* d_in, const int* in_sizes, int n_in,
                              void* d_out, int out_size, void* d_ws, size_t ws_size,
                              hipStream_t stream) {
    (void)in_sizes; (void)n_in; (void)out_size; (void)d_ws; (void)ws_size;
    const float* x = (const float*)d_in[0];
    float* out = (float*)d_out;
    // 43520 tiles total (4 levels x 128 batch), 8 waves (tiles) per block
    const int blocks = 43520 / WAVES_PER_BLOCK;   // 5440
    MultiscaleLLN_56367150793312_kernel<<<blocks, 256, 0, stream>>>(x, out);
}